// MultiHeadedSparseAttentionModule_27590869910179
// MI455X (gfx1250) — compile-verified
//
#include <hip/hip_runtime.h>

typedef __attribute__((ext_vector_type(2))) float v2f;
typedef __attribute__((ext_vector_type(4))) float v4f;
typedef __attribute__((ext_vector_type(8))) float v8f;

#define SWS 80   // padded row stride (in float2) for the paired-W LDS tile:
                 // +1 row = +160 dwords = +32 banks -> lanes 0-15 / 16-31 disjoint

// ------------------------------------------------------------------
// GEMM: Y[M x 64] = X[M x 64] @ W[64 x 64] + bias.
// 256 threads = 8 waves; each wave computes 16 rows x 64 cols with
// V_WMMA_F32_16X16X4_F32 (4 column-tile accumulators, 16 K-steps).
// W is staged in LDS pre-paired: sWp[k/2][n] = {W[k][n], W[k+1][n]},
// so every B operand is one contiguous, bank-conflict-free ds_load_b64.
// ------------------------------------------------------------------
__global__ __launch_bounds__(256) void gemm64_kernel(
    const float* __restrict__ X, const float* __restrict__ W,
    const float* __restrict__ bias, float* __restrict__ Y, int M)
{
    __shared__ v2f   sWp[32 * SWS];
    __shared__ float sB[64];
    int t = threadIdx.x;
#pragma unroll
    for (int i = t; i < 2048; i += 256) {
        int kp = i >> 6, n = i & 63;
        v2f w;
        w.x = W[(2 * kp)     * 64 + n];
        w.y = W[(2 * kp + 1) * 64 + n];
        sWp[kp * SWS + n] = w;
    }
    if (t < 64) sB[t] = bias[t];
    __syncthreads();

    int lane = t & 31;
    int wave = t >> 5;
    int half = lane >> 4;     // 0: K={k0,k0+1}   1: K={k0+2,k0+3}
    int l16  = lane & 15;
    int rowBase = blockIdx.x * 128 + wave * 16;
    int arow = rowBase + l16;
    int rclamp = arow < M ? arow : M - 1;      // clamp keeps EXEC full for WMMA
    const float* xp = X + (size_t)rclamp * 64;

    v8f acc0 = {}, acc1 = {}, acc2 = {}, acc3 = {};
#pragma unroll
    for (int k0 = 0; k0 < 64; k0 += 4) {
        int kk = k0 + 2 * half;
        v2f a; a.x = xp[kk]; a.y = xp[kk + 1];
        const v2f* wp = sWp + ((k0 >> 1) + half) * SWS + l16;
        v2f b0 = wp[0], b1 = wp[16], b2 = wp[32], b3 = wp[48];
        acc0 = __builtin_amdgcn_wmma_f32_16x16x4_f32(false, a, false, b0, (short)0, acc0, false, false);
        acc1 = __builtin_amdgcn_wmma_f32_16x16x4_f32(false, a, false, b1, (short)0, acc1, false, false);
        acc2 = __builtin_amdgcn_wmma_f32_16x16x4_f32(false, a, false, b2, (short)0, acc2, false, false);
        acc3 = __builtin_amdgcn_wmma_f32_16x16x4_f32(false, a, false, b3, (short)0, acc3, false, false);
    }

    float bb0 = sB[l16], bb1 = sB[l16 + 16], bb2 = sB[l16 + 32], bb3 = sB[l16 + 48];
#pragma unroll
    for (int r = 0; r < 8; ++r) {
        int orow = rowBase + r + 8 * half;
        if (orow < M) {
            float* yp = Y + (size_t)orow * 64 + l16;
            yp[0]  = acc0[r] + bb0;
            yp[16] = acc1[r] + bb1;
            yp[32] = acc2[r] + bb2;
            yp[48] = acc3[r] + bb3;
        }
    }
}

// ------------------------------------------------------------------
// Scaled GEMM for the oe branch: Y = (scores ⊙ X) @ W + bias,
// X[e, c] scaled by S[e, c & 7] (head = col % NH).
// Safe with Y == X: each wave finishes reading its own 16 rows before
// storing them, and no other wave touches those rows.
// ------------------------------------------------------------------
__global__ __launch_bounds__(256) void gemm64_scaled_kernel(
    const float* X, const float* __restrict__ S,
    const float* __restrict__ W, const float* __restrict__ bias,
    float* Y, int M)
{
    __shared__ v2f   sWp[32 * SWS];
    __shared__ float sB[64];
    int t = threadIdx.x;
#pragma unroll
    for (int i = t; i < 2048; i += 256) {
        int kp = i >> 6, n = i & 63;
        v2f w;
        w.x = W[(2 * kp)     * 64 + n];
        w.y = W[(2 * kp + 1) * 64 + n];
        sWp[kp * SWS + n] = w;
    }
    if (t < 64) sB[t] = bias[t];
    __syncthreads();

    int lane = t & 31;
    int wave = t >> 5;
    int half = lane >> 4;
    int l16  = lane & 15;
    int rowBase = blockIdx.x * 128 + wave * 16;
    int arow = rowBase + l16;
    int rclamp = arow < M ? arow : M - 1;
    const float* xp = X + (size_t)rclamp * 64;

    float sc[8];
#pragma unroll
    for (int h = 0; h < 8; ++h) sc[h] = S[(size_t)rclamp * 8 + h];

    v8f acc0 = {}, acc1 = {}, acc2 = {}, acc3 = {};
#pragma unroll
    for (int k0 = 0; k0 < 64; k0 += 4) {
        int kk = k0 + 2 * half;
        v2f a;
        a.x = xp[kk]     * sc[kk & 7];
        a.y = xp[kk + 1] * sc[(kk + 1) & 7];
        const v2f* wp = sWp + ((k0 >> 1) + half) * SWS + l16;
        v2f b0 = wp[0], b1 = wp[16], b2 = wp[32], b3 = wp[48];
        acc0 = __builtin_amdgcn_wmma_f32_16x16x4_f32(false, a, false, b0, (short)0, acc0, false, false);
        acc1 = __builtin_amdgcn_wmma_f32_16x16x4_f32(false, a, false, b1, (short)0, acc1, false, false);
        acc2 = __builtin_amdgcn_wmma_f32_16x16x4_f32(false, a, false, b2, (short)0, acc2, false, false);
        acc3 = __builtin_amdgcn_wmma_f32_16x16x4_f32(false, a, false, b3, (short)0, acc3, false, false);
    }

    float bb0 = sB[l16], bb1 = sB[l16 + 16], bb2 = sB[l16 + 32], bb3 = sB[l16 + 48];
#pragma unroll
    for (int r = 0; r < 8; ++r) {
        int orow = rowBase + r + 8 * half;
        if (orow < M) {
            float* yp = Y + (size_t)orow * 64 + l16;
            yp[0]  = acc0[r] + bb0;
            yp[16] = acc1[r] + bb1;
            yp[32] = acc2[r] + bb2;
            yp[48] = acc3[r] + bb3;
        }
    }
}

__device__ __forceinline__ void atomicMaxF(float* addr, float v)
{
    if (v >= 0.f) atomicMax((int*)addr, __float_as_int(v));
    else          atomicMin((unsigned int*)addr, __float_as_uint(v));
}

// ------------------------------------------------------------------
// SDDMM + edge gate + fused segment-max. One thread per (edge, head).
// scores[e,h] = (sum_d q[row,d,h]*k[col,d,h]) / sqrt(8) * sum_d e[e,d,h]
// ------------------------------------------------------------------
__global__ __launch_bounds__(256) void sddmm_kernel(
    const float* __restrict__ q, const float* __restrict__ k,
    const float* __restrict__ elin, const int* __restrict__ row,
    const int* __restrict__ col, float* __restrict__ scores,
    float* __restrict__ mx, int E_)
{
    int idx = blockIdx.x * 256 + threadIdx.x;
    int e = idx >> 3, h = idx & 7;
    if (e >= E_) return;
    int r = row[e], c = col[e];
    const float* qp = q + (size_t)r * 64 + h;
    const float* kp = k + (size_t)c * 64 + h;
    const float* ep = elin + (size_t)e * 64 + h;
    float dot = 0.f, gate = 0.f;
#pragma unroll
    for (int d = 0; d < 8; ++d) {
        dot  = fmaf(qp[d * 8], kp[d * 8], dot);
        gate += ep[d * 8];
    }
    float s = dot * 0.35355339059327373f * gate;
    scores[idx] = s;
    atomicMaxF(&mx[(size_t)r * 8 + h], s);
}

__global__ __launch_bounds__(256) void init_kernel(
    float* __restrict__ mx, float* __restrict__ den,
    float* __restrict__ hout, int n)
{
    int i = blockIdx.x * 256 + threadIdx.x;
    if (i < n * 8) { mx[i] = -__builtin_inff(); den[i] = 0.f; }
    if (i < n * 64) hout[i] = 0.f;
}

// ex = exp(scores - mx[row]); overwrite scores with ex; den += ex
__global__ __launch_bounds__(256) void expsum_kernel(
    float* __restrict__ scores, const int* __restrict__ row,
    const float* __restrict__ mx, float* __restrict__ den, int E_)
{
    int idx = blockIdx.x * 256 + threadIdx.x;
    int e = idx >> 3, h = idx & 7;
    if (e >= E_) return;
    size_t seg = (size_t)row[e] * 8 + h;
    float ex = __expf(scores[idx] - mx[seg]);
    scores[idx] = ex;
    atomicAdd(&den[seg], ex);
}

__global__ __launch_bounds__(256) void recip_kernel(float* __restrict__ den, int n8)
{
    int i = blockIdx.x * 256 + threadIdx.x;
    if (i < n8) {
        float d = den[i];
        den[i] = d > 0.f ? 1.f / d : 0.f;
    }
}

// Scatter bSpMM: hout[row, c] += sm[e, c&7] * v[col, c]
__global__ __launch_bounds__(256) void spmm_kernel(
    const float* __restrict__ ex, const float* __restrict__ rden,
    const float* __restrict__ v, const int* __restrict__ row,
    const int* __restrict__ col, float* __restrict__ hout, int E_)
{
    long long idx = (long long)blockIdx.x * 256 + threadIdx.x;
    int e = (int)(idx >> 6), c = (int)(idx & 63), h = c & 7;
    if (e >= E_) return;
    int r = row[e], cl = col[e];
    float sm = ex[(size_t)e * 8 + h] * rden[(size_t)r * 8 + h];
    atomicAdd(&hout[(size_t)r * 64 + c], sm * v[(size_t)cl * 64 + c]);
}

// ------------------------------------------------------------------
extern "C" void kernel_launch(void* const* d_in, const int* in_sizes, int n_in,
                              void* d_out, int out_size, void* d_ws, size_t ws_size,
                              hipStream_t stream)
{
    const float* h_x = (const float*)d_in[0];
    const float* h_e = (const float*)d_in[1];
    const int*   row = (const int*)d_in[2];
    const int*   col = (const int*)d_in[3];
    const float* Wq  = (const float*)d_in[4];  const float* bq  = (const float*)d_in[5];
    const float* Wk  = (const float*)d_in[6];  const float* bk  = (const float*)d_in[7];
    const float* Wv  = (const float*)d_in[8];  const float* bv  = (const float*)d_in[9];
    const float* We  = (const float*)d_in[10]; const float* be  = (const float*)d_in[11];
    const float* Woh = (const float*)d_in[12]; const float* boh = (const float*)d_in[13];
    const float* Woe = (const float*)d_in[14]; const float* boe = (const float*)d_in[15];

    const int N_ = in_sizes[0] / 64;
    const int E_ = in_sizes[2];

    float* out = (float*)d_out;
    float* oh  = out;                       // N x 64
    float* oe  = out + (size_t)N_ * 64;     // E x 64 ; doubles as elin staging

    float* ws     = (float*)d_ws;
    float* qlin   = ws;
    float* klin   = qlin + (size_t)N_ * 64;
    float* vlin   = klin + (size_t)N_ * 64;
    float* hout   = vlin + (size_t)N_ * 64;
    float* mx     = hout + (size_t)N_ * 64;
    float* den    = mx   + (size_t)N_ * 8;
    float* scores = den  + (size_t)N_ * 8;  // E x 8

    dim3 blk(256);
    int gN  = (N_ + 127) / 128;
    int gE  = (E_ + 127) / 128;
    int gEH = (int)(((long long)E_ * 8 + 255) / 256);
    int gS  = (int)(((long long)E_ * 64 + 255) / 256);

    // init segment buffers first (no dependencies; lets SDDMM fuse segment-max)
    init_kernel<<<(N_ * 64 + 255) / 256, blk, 0, stream>>>(mx, den, hout, N_);
    // node linears (WMMA)
    gemm64_kernel<<<gN, blk, 0, stream>>>(h_x, Wq, bq, qlin, N_);
    gemm64_kernel<<<gN, blk, 0, stream>>>(h_x, Wk, bk, klin, N_);
    gemm64_kernel<<<gN, blk, 0, stream>>>(h_x, Wv, bv, vlin, N_);
    // edge linear, staged into the oe output slot (WMMA)
    gemm64_kernel<<<gE, blk, 0, stream>>>(h_e, We, be, oe, E_);
    // SDDMM + gate + segment-max
    sddmm_kernel<<<gEH, blk, 0, stream>>>(qlin, klin, oe, row, col, scores, mx, E_);
    // oe = (scores ⊙ elin) @ Woe + boe, in place over the staging (WMMA)
    gemm64_scaled_kernel<<<gE, blk, 0, stream>>>(oe, scores, Woe, boe, oe, E_);
    // sparse softmax remainder
    expsum_kernel<<<gEH, blk, 0, stream>>>(scores, row, mx, den, E_);
    recip_kernel<<<(N_ * 8 + 255) / 256, blk, 0, stream>>>(den, N_ * 8);
    // weighted scatter aggregation
    spmm_kernel<<<gS, blk, 0, stream>>>(scores, den, vlin, row, col, hout, E_);
    // oh = hout @ Woh + boh (WMMA)
    gemm64_kernel<<<gN, blk, 0, stream>>>(hout, Woh, boh, oh, N_);
}